// ScattterAttentionLayer_69337952026835
// MI455X (gfx1250) — compile-verified
//
#include <hip/hip_runtime.h>
#include <math.h>

#define N_NODES 100000
#define E_EDGES 3200000
#define IN_F    128
#define OUT_F   64
#define ALPHA   0.1f

typedef __attribute__((ext_vector_type(2))) float v2f;
typedef __attribute__((ext_vector_type(8))) float v8f;

// ---------------------------------------------------------------------------
// Dense GEMM: support0[N,64] = x[N,128] @ W[128,64] via V_WMMA_F32_16X16X4_F32.
// One wave per 16x64 output STRIP: 4 accumulators (tileN=0..3), so each A
// fragment is loaded once and feeds 4 independent WMMAs (no D->A hazards,
// XDL stays busy). W (32 KB) is staged in LDS once per 256-thread block.
// N = 6250*16 exactly; OUT_F = 4*16 exactly -> no ragged tiles.
//
// Layouts (ISA 7.12.2):
//   A 16x4 f32 : lane%16 = M row; lanes 0-15 hold K={0,1}, lanes 16-31 K={2,3}.
//   B 4x16 f32 : lane%16 = N col; lane/16 picks K pair, the 2 per-lane VGPRs
//                hold the two K rows.
//   C/D 16x16  : VGPR p -> row p + 8*(lane/16), col = lane%16.
// ---------------------------------------------------------------------------
__global__ __launch_bounds__(256) void gemm_xw_wmma(const float* __restrict__ x,
                                                    const float* __restrict__ W,
                                                    float* __restrict__ out) {
    __shared__ float Wlds[IN_F * OUT_F];          // 32 KB (of 320 KB/WGP)

    // Cooperative W load: 8192 floats, 256 threads, float4 -> 8 iters/thread.
    for (int idx = (int)threadIdx.x * 4; idx < IN_F * OUT_F; idx += 256 * 4)
        *(float4*)&Wlds[idx] = *(const float4*)&W[idx];
    __syncthreads();

    const int lane  = (int)(threadIdx.x & 31);
    const int tileM = (int)blockIdx.x * 8 + (int)(threadIdx.x >> 5);
    if (tileM >= N_NODES / 16) return;            // after the only barrier: legal

    const int m     = (lane & 15);
    const int khalf = (lane >> 4);                // 0: K={0,1}, 1: K={2,3}
    const int mrow  = tileM * 16 + m;

    v8f c0 = {}, c1 = {}, c2 = {}, c3 = {};
#pragma unroll 4
    for (int k0 = 0; k0 < IN_F; k0 += 4) {
        v2f a;
        a.x = x[(size_t)mrow * IN_F + k0 + khalf * 2 + 0];
        a.y = x[(size_t)mrow * IN_F + k0 + khalf * 2 + 1];

        const float* wr0 = &Wlds[(k0 + khalf * 2 + 0) * OUT_F + m];
        const float* wr1 = &Wlds[(k0 + khalf * 2 + 1) * OUT_F + m];
        v2f b0, b1, b2, b3;
        b0.x = wr0[0];  b0.y = wr1[0];
        b1.x = wr0[16]; b1.y = wr1[16];
        b2.x = wr0[32]; b2.y = wr1[32];
        b3.x = wr0[48]; b3.y = wr1[48];

        c0 = __builtin_amdgcn_wmma_f32_16x16x4_f32(false, a, false, b0, (short)0, c0, false, false);
        c1 = __builtin_amdgcn_wmma_f32_16x16x4_f32(false, a, false, b1, (short)0, c1, false, false);
        c2 = __builtin_amdgcn_wmma_f32_16x16x4_f32(false, a, false, b2, (short)0, c2, false, false);
        c3 = __builtin_amdgcn_wmma_f32_16x16x4_f32(false, a, false, b3, (short)0, c3, false, false);
    }

    const int rbase = tileM * 16 + khalf * 8;
#pragma unroll
    for (int p = 0; p < 8; ++p) {
        float* o = out + (size_t)(rbase + p) * OUT_F + m;
        o[0]  = c0[p];
        o[16] = c1[p];
        o[32] = c2[p];
        o[48] = c3[p];
    }
}

// ---------------------------------------------------------------------------
// Zero-fill (graph-capture safe; harness poisons d_ws with 0xAA).
// ---------------------------------------------------------------------------
__global__ void zero_f32(float* __restrict__ p, size_t n) {
    size_t i = (size_t)blockIdx.x * blockDim.x + threadIdx.x;
    const size_t stride = (size_t)gridDim.x * blockDim.x;
    for (; i < n; i += stride) p[i] = 0.0f;
}

// ---------------------------------------------------------------------------
// SpMM scatter: Y[rows[e], :] += vals[e] * X[cols[e], :]
// One wave32 per edge; lane l handles features 2l, 2l+1 (float2 gather, two
// lane-consecutive global_atomic_add_f32 -> coalesced L2 atomic packets).
// X and Y (25.6 MB each) stay resident in the 192 MB L2.
// ---------------------------------------------------------------------------
__global__ __launch_bounds__(256) void spmm_scatter(const int*   __restrict__ rows,
                                                    const int*   __restrict__ cols,
                                                    const float* __restrict__ vals,
                                                    const float* __restrict__ X,
                                                    float*       __restrict__ Y) {
    const int e    = (int)blockIdx.x * 8 + (int)(threadIdx.x >> 5);
    const int lane = (int)(threadIdx.x & 31);
    if (e >= E_EDGES) return;

    const int   r = rows[e];            // wave-uniform
    const int   s = cols[e];
    const float v = vals[e];

    const float2 xv = ((const float2*)(X + (size_t)s * OUT_F))[lane];
    float* yp = Y + (size_t)r * OUT_F + 2 * lane;
    atomicAdd(yp + 0, v * xv.x);
    atomicAdd(yp + 1, v * xv.y);
}

// ---------------------------------------------------------------------------
// In-place |x| over the three P-path slabs.
// ---------------------------------------------------------------------------
__global__ void abs_inplace(float* __restrict__ p, size_t n) {
    size_t i = (size_t)blockIdx.x * blockDim.x + threadIdx.x;
    const size_t stride = (size_t)gridDim.x * blockDim.x;
    for (; i < n; i += stride) p[i] = fabsf(p[i]);
}

// ---------------------------------------------------------------------------
// Attention: one wave32 per node (torch raw-view semantics preserved):
//   i <  N/2: a_input row k = [h[2i] ; h[2i+1]]          (same for all k)
//   i >= N/2: a_input row k = [Hk[2i-N] ; Hk[2i-N+1]]
//   e = leaky_relu(dot), att = softmax(e),
//   h_prime[i,f] = (1/6) sum_j att[j] * H_{(j*64+f)%6}[i, (j*64+f)/6]
// H slabs contiguous: Hbase + k*slab (slab select is arithmetic, no spills).
// ---------------------------------------------------------------------------
__global__ __launch_bounds__(256) void attention_combine(
        const float* __restrict__ h,       // support0 [N,64]
        const float* __restrict__ Hbase,   // 6 slabs of [N,64]
        const float* __restrict__ a,       // [128]
        float* __restrict__ h_prime,       // [N,64]
        float* __restrict__ att_out) {     // [N,6]
    const int i    = (int)blockIdx.x * 8 + (int)(threadIdx.x >> 5);
    const int lane = (int)(threadIdx.x & 31);
    if (i >= N_NODES) return;

    const size_t slab = (size_t)N_NODES * OUT_F;
    const float a0 = a[lane], a1 = a[lane + 32], a2 = a[lane + 64], a3 = a[lane + 96];

    const bool   firstHalf = (i < N_NODES / 2);
    const size_t r0 = firstHalf ? (size_t)(2 * i) : (size_t)(2 * i - N_NODES);

    float e[6];
#pragma unroll
    for (int k = 0; k < 6; ++k) {
        const float* src = firstHalf ? h : (Hbase + (size_t)k * slab);
        float part = src[r0 * OUT_F + lane]            * a0
                   + src[r0 * OUT_F + lane + 32]       * a1
                   + src[(r0 + 1) * OUT_F + lane]      * a2
                   + src[(r0 + 1) * OUT_F + lane + 32] * a3;
#pragma unroll
        for (int off = 16; off > 0; off >>= 1)      // wave32 butterfly reduce
            part += __shfl_xor(part, off, 32);
        e[k] = part;                                 // all lanes hold full dot
    }

    float mx = -INFINITY;
#pragma unroll
    for (int k = 0; k < 6; ++k) {
        e[k] = e[k] > 0.0f ? e[k] : ALPHA * e[k];    // leaky relu
        mx = fmaxf(mx, e[k]);
    }
    float w[6], sum = 0.0f;
#pragma unroll
    for (int k = 0; k < 6; ++k) { w[k] = __expf(e[k] - mx); sum += w[k]; }
    const float inv = 1.0f / sum;
#pragma unroll
    for (int k = 0; k < 6; ++k) w[k] *= inv;

    if (lane == 0) {
#pragma unroll
        for (int k = 0; k < 6; ++k) att_out[(size_t)i * 6 + k] = w[k];
    }

#pragma unroll
    for (int t = 0; t < 2; ++t) {
        const int f = lane + t * 32;
        float acc = 0.0f;
#pragma unroll
        for (int j = 0; j < 6; ++j) {
            const int idx = j * 64 + f;
            const int kk  = idx % 6;                 // which H slab (raw view)
            const int fp  = idx / 6;                 // feature within it
            acc += w[j] * Hbase[(size_t)kk * slab + (size_t)i * OUT_F + fp];
        }
        h_prime[(size_t)i * OUT_F + f] = acc * (1.0f / 6.0f);
    }
}

// ---------------------------------------------------------------------------
// kernel_launch
// d_in order: x, W, a, A_{r,c,v}, P1_{r,c,v}, P2_{r,c,v}, P3_{r,c,v}
// d_out: h_prime (N*64 floats) ++ attention (N*6 floats)
// d_ws layout: [support0 : N*64] [H slabs : 6 * N*64]  -> 179.2 MB
// ---------------------------------------------------------------------------
extern "C" void kernel_launch(void* const* d_in, const int* in_sizes, int n_in,
                              void* d_out, int out_size, void* d_ws, size_t ws_size,
                              hipStream_t stream) {
    const float* x   = (const float*)d_in[0];
    const float* W   = (const float*)d_in[1];
    const float* a   = (const float*)d_in[2];
    const int*   Ar  = (const int*)d_in[3];
    const int*   Ac  = (const int*)d_in[4];
    const float* Av  = (const float*)d_in[5];
    const int*   P1r = (const int*)d_in[6];
    const int*   P1c = (const int*)d_in[7];
    const float* P1v = (const float*)d_in[8];
    const int*   P2r = (const int*)d_in[9];
    const int*   P2c = (const int*)d_in[10];
    const float* P2v = (const float*)d_in[11];
    const int*   P3r = (const int*)d_in[12];
    const int*   P3c = (const int*)d_in[13];
    const float* P3v = (const float*)d_in[14];

    const size_t slab = (size_t)N_NODES * OUT_F;
    float* support0 = (float*)d_ws;
    float* H        = support0 + slab;               // 6 contiguous slabs

    float* h_prime = (float*)d_out;
    float* att_out = (float*)d_out + slab;

    // 1) support0 = x @ W  (wave per 16x64 strip; 6250 strips, 8 waves/block)
    gemm_xw_wmma<<<(N_NODES / 16 + 7) / 8, 256, 0, stream>>>(x, W, support0);

    // 2) zero the six scatter targets
    zero_f32<<<2048, 256, 0, stream>>>(H, 6 * slab);

    // 3) six SpMMs (wave-per-edge, E/8 blocks of 256)
    const int spmm_blocks = E_EDGES / 8;
    spmm_scatter<<<spmm_blocks, 256, 0, stream>>>(Ar,  Ac,  Av,  support0,     H + 0 * slab); // h_A
    spmm_scatter<<<spmm_blocks, 256, 0, stream>>>(Ar,  Ac,  Av,  H + 0 * slab, H + 1 * slab); // h_A2
    spmm_scatter<<<spmm_blocks, 256, 0, stream>>>(Ar,  Ac,  Av,  H + 1 * slab, H + 2 * slab); // h_A3
    spmm_scatter<<<spmm_blocks, 256, 0, stream>>>(P1r, P1c, P1v, support0,     H + 3 * slab);
    spmm_scatter<<<spmm_blocks, 256, 0, stream>>>(P2r, P2c, P2v, support0,     H + 4 * slab);
    spmm_scatter<<<spmm_blocks, 256, 0, stream>>>(P3r, P3c, P3v, support0,     H + 5 * slab);

    // 4) abs() over the three P slabs
    abs_inplace<<<2048, 256, 0, stream>>>(H + 3 * slab, 3 * slab);

    // 5) attention + combine (wave-per-node)
    attention_combine<<<N_NODES / 8, 256, 0, stream>>>(support0, H, a, h_prime, att_out);
}